// MMD_loss_20315195310689
// MI455X (gfx1250) — compile-verified
//
#include <hip/hip_runtime.h>

// MMD loss, fused: Gram matrix via V_WMMA_F32_16X16X4_F32, bandwidth via the
// algebraic identity sum(L2) = 2n*sum(|x|^2) - 2*|sum(x)|^2, and the 5-kernel
// Gaussian sum + block-weighted reduction fused into the GEMM epilogue.
// The 5 kernel widths are a geometric ladder (bw*2^i), so the epilogue uses
// ONE v_exp_f32 plus 4 squarings per pair: t=exp2(e/16), then t^2,t^4,t^8,t^16.
// All reductions are fixed-order (no float atomics) => deterministic output.

typedef float v2f __attribute__((ext_vector_type(2)));
typedef float v8f __attribute__((ext_vector_type(8)));

#define NS   4096      // rows in source (= rows in target)
#define NTOT 8192      // total rows
#define D    256       // feature dim (K of the Gram GEMM)
#define TLDS 36        // LDS row stride in floats: 144B = 16B-aligned, conflict-free

// workspace layout (in floats)
#define WS_SQ      0        // 8192 : per-row squared norms
#define WS_COLPART 8192     // 32*256 : per-block partial column sums
#define WS_BW      16384    // 1 : bandwidth
#define WS_PARTIAL 16640    // 4096 : per-tile weighted partial sums

// ---------------------------------------------------------------- row |x|^2
__global__ __launch_bounds__(256)
void mmd_rowsq(const float* __restrict__ src, const float* __restrict__ tgt,
               float* __restrict__ sq) {
  const int tid  = threadIdx.x;
  const int lane = tid & 31;
  const int row  = blockIdx.x * 8 + (tid >> 5);   // one wave32 per row
  const float* rp = (row < NS) ? (src + (size_t)row * D)
                               : (tgt + (size_t)(row - NS) * D);
  float s = 0.f;
#pragma unroll
  for (int t = 0; t < 8; ++t) {
    float v = rp[lane + 32 * t];
    s = fmaf(v, v, s);
  }
#pragma unroll
  for (int o = 16; o > 0; o >>= 1) s += __shfl_down(s, o, 32);
  if (lane == 0) sq[row] = s;
}

// ------------------------------------------------- partial column sums (32 blocks)
__global__ __launch_bounds__(256)
void mmd_colpart(const float* __restrict__ src, const float* __restrict__ tgt,
                 float* __restrict__ colpart) {
  const int tid = threadIdx.x;
  const int r0  = blockIdx.x * 256;               // 256 | 4096 => whole block same side
  const float* base = (r0 < NS) ? (src + (size_t)r0 * D)
                                : (tgt + (size_t)(r0 - NS) * D);
  float s = 0.f;
  for (int i = 0; i < 256; ++i) s += base[(size_t)i * D + tid];
  colpart[blockIdx.x * 256 + tid] = s;
}

// --------------------------------------------------------- bandwidth (1 block)
__global__ __launch_bounds__(256)
void mmd_bw(const float* __restrict__ sq, const float* __restrict__ colpart,
            float* __restrict__ bw) {
  __shared__ float red[256];
  __shared__ float snorm_sh;
  const int tid = threadIdx.x;

  float s = 0.f;
#pragma unroll
  for (int b = 0; b < 32; ++b) s += colpart[b * 256 + tid];
  red[tid] = s * s;                                // -> |sum_i x_i|^2
  __syncthreads();
  for (int o = 128; o > 0; o >>= 1) {
    if (tid < o) red[tid] += red[tid + o];
    __syncthreads();
  }
  if (tid == 0) snorm_sh = red[0];
  __syncthreads();

  float q = 0.f;
#pragma unroll
  for (int r = 0; r < 32; ++r) q += sq[tid + 256 * r];
  red[tid] = q;                                    // -> sum_i |x_i|^2
  __syncthreads();
  for (int o = 128; o > 0; o >>= 1) {
    if (tid < o) red[tid] += red[tid + o];
    __syncthreads();
  }
  if (tid == 0) {
    const float n = (float)NTOT;
    const float sumL2 = 2.f * n * red[0] - 2.f * snorm_sh;
    bw[0] = (sumL2 / (n * n - n)) * 0.25f;         // / KERNEL_MUL^(KERNEL_NUM//2)
  }
}

// ------------------------------------- fused Gram tile + Gaussian-kernel epilogue
__global__ __launch_bounds__(256)
void mmd_tile(const float* __restrict__ src, const float* __restrict__ tgt,
              const float* __restrict__ sq, const float* __restrict__ bwp,
              float* __restrict__ partial) {
  __shared__ float As[128 * TLDS];
  __shared__ float Bs[128 * TLDS];
  __shared__ float sqI[128];
  __shared__ float sqJ[128];
  __shared__ float red[256];

  const int tid  = threadIdx.x;
  const int lane = tid & 31;
  const int wav  = tid >> 5;     // 8 waves
  const int wm   = wav & 3;      // wave row  (32 rows each)
  const int wn   = wav >> 2;     // wave col  (64 cols each)
  const int half = lane >> 4;    // K-pair select per WMMA f32 layout
  const int lr   = lane & 15;    // M (A) / N (B,C) index

  const int i0 = blockIdx.x * 128;
  const int j0 = blockIdx.y * 128;
  const float* Xi = (i0 < NS) ? (src + (size_t)i0 * D) : (tgt + (size_t)(i0 - NS) * D);
  const float* Xj = (j0 < NS) ? (src + (size_t)j0 * D) : (tgt + (size_t)(j0 - NS) * D);

  if (tid < 128) sqI[tid] = sq[i0 + tid];
  else           sqJ[tid - 128] = sq[j0 + (tid - 128)];

  v8f zero;
#pragma unroll
  for (int e = 0; e < 8; ++e) zero[e] = 0.f;
  v8f acc[2][4];
#pragma unroll
  for (int m = 0; m < 2; ++m)
#pragma unroll
    for (int nf = 0; nf < 4; ++nf) acc[m][nf] = zero;

  for (int kc = 0; kc < D; kc += 32) {
    __syncthreads();
#pragma unroll
    for (int r = 0; r < 4; ++r) {                  // 128x32 chunk of A and B
      const int q   = tid + 256 * r;
      const int row = q >> 3;
      const int c4  = q & 7;
      const float4 av = *reinterpret_cast<const float4*>(Xi + (size_t)row * D + kc + c4 * 4);
      const float4 bv = *reinterpret_cast<const float4*>(Xj + (size_t)row * D + kc + c4 * 4);
      *reinterpret_cast<float4*>(&As[row * TLDS + c4 * 4]) = av;
      *reinterpret_cast<float4*>(&Bs[row * TLDS + c4 * 4]) = bv;
    }
    __syncthreads();
#pragma unroll
    for (int k = 0; k < 32; k += 4) {
      const int kidx = k + half * 2;
      const v2f a0 = *reinterpret_cast<const v2f*>(&As[(wm * 32 +  0 + lr) * TLDS + kidx]);
      const v2f a1 = *reinterpret_cast<const v2f*>(&As[(wm * 32 + 16 + lr) * TLDS + kidx]);
      const v2f b0 = *reinterpret_cast<const v2f*>(&Bs[(wn * 64 +  0 + lr) * TLDS + kidx]);
      const v2f b1 = *reinterpret_cast<const v2f*>(&Bs[(wn * 64 + 16 + lr) * TLDS + kidx]);
      const v2f b2 = *reinterpret_cast<const v2f*>(&Bs[(wn * 64 + 32 + lr) * TLDS + kidx]);
      const v2f b3 = *reinterpret_cast<const v2f*>(&Bs[(wn * 64 + 48 + lr) * TLDS + kidx]);
      acc[0][0] = __builtin_amdgcn_wmma_f32_16x16x4_f32(false, a0, false, b0, (short)0, acc[0][0], false, false);
      acc[0][1] = __builtin_amdgcn_wmma_f32_16x16x4_f32(false, a0, false, b1, (short)0, acc[0][1], false, false);
      acc[0][2] = __builtin_amdgcn_wmma_f32_16x16x4_f32(false, a0, false, b2, (short)0, acc[0][2], false, false);
      acc[0][3] = __builtin_amdgcn_wmma_f32_16x16x4_f32(false, a0, false, b3, (short)0, acc[0][3], false, false);
      acc[1][0] = __builtin_amdgcn_wmma_f32_16x16x4_f32(false, a1, false, b0, (short)0, acc[1][0], false, false);
      acc[1][1] = __builtin_amdgcn_wmma_f32_16x16x4_f32(false, a1, false, b1, (short)0, acc[1][1], false, false);
      acc[1][2] = __builtin_amdgcn_wmma_f32_16x16x4_f32(false, a1, false, b2, (short)0, acc[1][2], false, false);
      acc[1][3] = __builtin_amdgcn_wmma_f32_16x16x4_f32(false, a1, false, b3, (short)0, acc[1][3], false, false);
    }
  }

  // epilogue: L2 -> 5 Gaussian kernels via one exp + 4 squarings -> local sum
  //   e = -L2/bw * log2(e);  t = exp2(e/16)
  //   kernels sum = t + t^2 + t^4 + t^8 + t^16
  //              = exp(-L2/(16bw)) + exp(-L2/(8bw)) + ... + exp(-L2/bw)
  const float bwv = bwp[0];
  const float e16s = -1.4426950408889634f / (16.f * bwv);  // exp2 scale for e/16
  float lsum = 0.f;
#pragma unroll
  for (int m = 0; m < 2; ++m) {
#pragma unroll
    for (int nf = 0; nf < 4; ++nf) {
#pragma unroll
      for (int v = 0; v < 8; ++v) {
        const int gm = wm * 32 + m * 16 + half * 8 + v;   // C/D layout: M = v + 8*half
        const int gn = wn * 64 + nf * 16 + lr;            // N = lane%16
        const float g  = acc[m][nf][v];
        const float L2 = fmaxf(sqI[gm] + sqJ[gn] - 2.f * g, 0.f);
        const float t1  = __builtin_amdgcn_exp2f(L2 * e16s); // exp(-L2/(bw*16))
        const float t2  = t1 * t1;                           // exp(-L2/(bw*8))
        const float t4  = t2 * t2;                           // exp(-L2/(bw*4))
        const float t8  = t4 * t4;                           // exp(-L2/(bw*2))
        const float t16 = t8 * t8;                           // exp(-L2/bw)
        lsum += t1 + t2 + t4 + t8 + t16;
      }
    }
  }

  red[tid] = lsum;
  __syncthreads();
  for (int o = 128; o > 0; o >>= 1) {
    if (tid < o) red[tid] += red[tid + o];
    __syncthreads();
  }
  if (tid == 0) {
    const float wgt = ((i0 < NS) == (j0 < NS)) ? 1.f : -1.f;  // SS/TT:+1, ST/TS:-1
    partial[blockIdx.y * gridDim.x + blockIdx.x] = red[0] * wgt;
  }
}

// ------------------------------------------------------- final reduce (1 block)
__global__ __launch_bounds__(256)
void mmd_final(const float* __restrict__ partial, float* __restrict__ out) {
  __shared__ float red[256];
  const int tid = threadIdx.x;
  float s = 0.f;
#pragma unroll
  for (int r = 0; r < 16; ++r) s += partial[tid + 256 * r];
  red[tid] = s;
  __syncthreads();
  for (int o = 128; o > 0; o >>= 1) {
    if (tid < o) red[tid] += red[tid + o];
    __syncthreads();
  }
  if (tid == 0) out[0] = red[0] * (1.f / ((float)NS * (float)NS));
}

extern "C" void kernel_launch(void* const* d_in, const int* in_sizes, int n_in,
                              void* d_out, int out_size, void* d_ws, size_t ws_size,
                              hipStream_t stream) {
  const float* src = (const float*)d_in[0];
  const float* tgt = (const float*)d_in[1];
  float* ws      = (float*)d_ws;
  float* sq      = ws + WS_SQ;
  float* colpart = ws + WS_COLPART;
  float* bw      = ws + WS_BW;
  float* partial = ws + WS_PARTIAL;
  float* outp    = (float*)d_out;

  mmd_rowsq  <<<dim3(NTOT / 8), dim3(256), 0, stream>>>(src, tgt, sq);
  mmd_colpart<<<dim3(32),       dim3(256), 0, stream>>>(src, tgt, colpart);
  mmd_bw     <<<dim3(1),        dim3(256), 0, stream>>>(sq, colpart, bw);
  mmd_tile   <<<dim3(64, 64),   dim3(256), 0, stream>>>(src, tgt, sq, bw, partial);
  mmd_final  <<<dim3(1),        dim3(256), 0, stream>>>(partial, outp);
}